// GCN_29231547416619
// MI455X (gfx1250) — compile-verified
//
#include <hip/hip_runtime.h>
#include <hip/hip_bf16.h>

typedef __attribute__((ext_vector_type(2))) float v2f;
typedef __attribute__((ext_vector_type(8))) float v8f;

#define N_NODES 100000
#define N_EDGES 1600000

// ---------------------------------------------------------------------------
// Degree / norm kernels
// ---------------------------------------------------------------------------
__global__ void deg_init_kernel(float* __restrict__ deg, int n) {
    int i = blockIdx.x * blockDim.x + threadIdx.x;
    if (i < n) deg[i] = 1.0f;  // self-loop contributes 1
}

__global__ void deg_accum_kernel(const int* __restrict__ dst, float* __restrict__ deg, int e) {
    int i = blockIdx.x * blockDim.x + threadIdx.x;
    if (i < e) atomicAdd(&deg[dst[i]], 1.0f);
}

__global__ void dinv_kernel(float* __restrict__ deg, int n) {
    int i = blockIdx.x * blockDim.x + threadIdx.x;
    if (i < n) deg[i] = rsqrtf(deg[i]);  // deg >= 1 always (self loop)
}

// ---------------------------------------------------------------------------
// fp32 WMMA GEMM: C[N x F] = A[N x K] * B[K x F]
// One block handles 16 consecutive rows. A tile staged in LDS (coalesced,
// shared by all waves). Each wave owns a 16-wide column strip.
// Wave32 layouts per CDNA5 ISA 7.12.2:
//   A 16x4 :  M = lane%16 ; VGPR0/1 hold K = {0,1} (lanes 0-15) / {2,3} (16-31)
//   B 4x16 :  N = lane%16 ; VGPR0/1 hold K = {0,1} (lanes 0-15) / {2,3} (16-31)
//   C 16x16: N = lane%16 ; VGPR r holds M = r (lanes 0-15) / 8+r (lanes 16-31)
// ---------------------------------------------------------------------------
template <int K>
__global__ void gemm16_wmma_kernel(const float* __restrict__ A,
                                   const float* __restrict__ B,
                                   float* __restrict__ C, int F) {
    __shared__ float As[16 * K];
    const int tid = threadIdx.x;
    const size_t row0 = (size_t)blockIdx.x * 16;

    // Stage 16 contiguous rows of A -> LDS (rows are contiguous in memory)
    const float* Ablk = A + row0 * K;
    for (int i = tid; i < 16 * K; i += blockDim.x) As[i] = Ablk[i];
    __syncthreads();

    const int wave = tid >> 5;
    const int lane = tid & 31;
    const int n    = lane & 15;   // N index (B/C/D), also M index for A loads
    const int half = lane >> 4;   // 0: K 0..1 / M 0..7 ; 1: K 2..3 / M 8..15
    const int col  = wave * 16 + n;

    v8f acc = {};
    for (int k = 0; k < K; k += 4) {
        const int ka = k + half * 2;
        v2f a, b;
        a.x = As[n * K + ka];
        a.y = As[n * K + ka + 1];
        b.x = B[(size_t)ka * F + col];
        b.y = B[(size_t)(ka + 1) * F + col];
        acc = __builtin_amdgcn_wmma_f32_16x16x4_f32(
            /*neg_a=*/false, a, /*neg_b=*/false, b,
            /*c_mod=*/(short)0, acc, /*reuse_a=*/false, /*reuse_b=*/false);
    }

#pragma unroll
    for (int r = 0; r < 8; ++r) {
        size_t row = row0 + (size_t)(half * 8 + r);
        C[row * F + col] = acc[r];
    }
}

// ---------------------------------------------------------------------------
// out[i,f] = h[i,f] * dinv[i]^2   (self-loop term, also initializes buffer)
// F fixed at 128.
// ---------------------------------------------------------------------------
__global__ void agg_init_kernel(const float* __restrict__ h,
                                const float* __restrict__ dinv,
                                float* __restrict__ out, size_t total) {
    size_t i = (size_t)blockIdx.x * blockDim.x + threadIdx.x;
    if (i < total) {
        float di = dinv[i >> 7];
        out[i] = h[i] * di * di;
    }
}

// ---------------------------------------------------------------------------
// Edge scatter: out[dst] += h[src] * dinv[src]*dinv[dst]
// One wave32 per edge; each lane moves 4 of the 128 features (float4 load +
// 4 global_atomic_add_f32).
// ---------------------------------------------------------------------------
__global__ void scatter128_kernel(const float* __restrict__ h,
                                  const int* __restrict__ src,
                                  const int* __restrict__ dst,
                                  const float* __restrict__ dinv,
                                  float* __restrict__ out, int e) {
    long gid = (long)blockIdx.x * blockDim.x + threadIdx.x;
    int edge = (int)(gid >> 5);
    int lane = (int)(gid & 31);
    if (edge >= e) return;
    int s = src[edge];
    int d = dst[edge];
    float nrm = dinv[s] * dinv[d];
    float4 v = ((const float4*)(h + (size_t)s * 128))[lane];
    float* o = out + (size_t)d * 128 + lane * 4;
    atomicAdd(o + 0, v.x * nrm);
    atomicAdd(o + 1, v.y * nrm);
    atomicAdd(o + 2, v.z * nrm);
    atomicAdd(o + 3, v.w * nrm);
}

// ---------------------------------------------------------------------------
// y[i] = (y[i] + b[i & mask]) with optional ReLU.  F is a power of two.
// ---------------------------------------------------------------------------
template <bool RELU>
__global__ void bias_act_kernel(float* __restrict__ y, const float* __restrict__ b,
                                size_t total, int mask) {
    size_t i = (size_t)blockIdx.x * blockDim.x + threadIdx.x;
    if (i < total) {
        float v = y[i] + b[i & mask];
        y[i] = RELU ? fmaxf(v, 0.0f) : v;
    }
}

// ---------------------------------------------------------------------------
// Softmax + argmax over 64 logits per node. One wave per node; each lane
// owns 2 columns. Wave32 shuffle reductions. Argmax = first max occurrence.
// ---------------------------------------------------------------------------
__global__ void softmax_argmax_kernel(const float* __restrict__ logits,
                                      float* __restrict__ soft,
                                      float* __restrict__ hard, int n) {
    long gid = (long)blockIdx.x * blockDim.x + threadIdx.x;
    int node = (int)(gid >> 5);
    int lane = (int)(gid & 31);
    if (node >= n) return;

    const float* l = logits + (size_t)node * 64;
    float v0 = l[lane * 2 + 0];
    float v1 = l[lane * 2 + 1];

    float m = fmaxf(v0, v1);
#pragma unroll
    for (int off = 16; off; off >>= 1) m = fmaxf(m, __shfl_xor(m, off, 32));

    float e0 = expf(v0 - m);
    float e1 = expf(v1 - m);
    float s = e0 + e1;
#pragma unroll
    for (int off = 16; off; off >>= 1) s += __shfl_xor(s, off, 32);
    float inv = 1.0f / s;

    soft[(size_t)node * 64 + lane * 2 + 0] = e0 * inv;
    soft[(size_t)node * 64 + lane * 2 + 1] = e1 * inv;

    float bv;
    int bi;
    if (v1 > v0) { bv = v1; bi = lane * 2 + 1; }
    else         { bv = v0; bi = lane * 2 + 0; }
#pragma unroll
    for (int off = 16; off; off >>= 1) {
        float ov = __shfl_xor(bv, off, 32);
        int   oi = __shfl_xor(bi, off, 32);
        if (ov > bv || (ov == bv && oi < bi)) { bv = ov; bi = oi; }
    }
    if (lane == 0) hard[node] = (float)bi;
}

// ---------------------------------------------------------------------------
// Launcher
// ---------------------------------------------------------------------------
extern "C" void kernel_launch(void* const* d_in, const int* in_sizes, int n_in,
                              void* d_out, int out_size, void* d_ws, size_t ws_size,
                              hipStream_t stream) {
    const float* x  = (const float*)d_in[0];   // [N,256]
    const int*   ei = (const int*)d_in[1];     // [2,E]
    const float* W1 = (const float*)d_in[2];   // [256,128]
    const float* b1 = (const float*)d_in[3];   // [128]
    const float* W2 = (const float*)d_in[4];   // [128,128]
    const float* b2 = (const float*)d_in[5];   // [128]
    const float* Wc = (const float*)d_in[6];   // [128,64]
    const float* bc = (const float*)d_in[7];   // [64]

    const int N = in_sizes[0] / 256;
    const int E = in_sizes[1] / 2;
    const int* src = ei;
    const int* dst = ei + E;

    // Output layout: logits [N*64] | emb [N*128] | soft [N*64] | hard [N]
    float* logits = (float*)d_out;
    float* emb    = logits + (size_t)N * 64;
    float* soft   = emb + (size_t)N * 128;
    float* hard   = soft + (size_t)N * 64;

    // Workspace: dinv [Npad] | buf1 [N*128] | buf2 [N*128]
    float* ws_f  = (float*)d_ws;
    size_t npad  = ((size_t)N + 255) & ~(size_t)255;
    float* dinv  = ws_f;
    float* buf1  = ws_f + npad;
    float* buf2  = buf1 + (size_t)N * 128;

    const size_t NF = (size_t)N * 128;
    const int nb_N   = (N + 255) / 256;
    const int nb_E   = (E + 255) / 256;
    const int nb_NF  = (int)((NF + 255) / 256);
    const int nb_Ew  = (int)(((long)E * 32 + 255) / 256);   // wave-per-edge
    const int nb_Nw  = (int)(((long)N * 32 + 255) / 256);   // wave-per-node
    const int nb_gem = N / 16;                              // 100000/16 = 6250

    // ---- degree + symmetric norm -------------------------------------------
    deg_init_kernel<<<nb_N, 256, 0, stream>>>(dinv, N);
    deg_accum_kernel<<<nb_E, 256, 0, stream>>>(dst, dinv, E);
    dinv_kernel<<<nb_N, 256, 0, stream>>>(dinv, N);

    // ---- layer 1: h = x @ W1 ; agg ; +b1 ; relu ----------------------------
    gemm16_wmma_kernel<256><<<nb_gem, 256, 0, stream>>>(x, W1, buf1, 128);
    agg_init_kernel<<<nb_NF, 256, 0, stream>>>(buf1, dinv, buf2, NF);
    scatter128_kernel<<<nb_Ew, 256, 0, stream>>>(buf1, src, dst, dinv, buf2, E);
    bias_act_kernel<true><<<nb_NF, 256, 0, stream>>>(buf2, b1, NF, 127);

    // ---- layer 2: h = a1 @ W2 ; agg (into emb output) ; +b2 ----------------
    gemm16_wmma_kernel<128><<<nb_gem, 256, 0, stream>>>(buf2, W2, buf1, 128);
    agg_init_kernel<<<nb_NF, 256, 0, stream>>>(buf1, dinv, emb, NF);
    scatter128_kernel<<<nb_Ew, 256, 0, stream>>>(buf1, src, dst, dinv, emb, E);
    bias_act_kernel<false><<<nb_NF, 256, 0, stream>>>(emb, b2, NF, 127);

    // ---- classifier: logits = emb @ Wc + bc ; softmax ; argmax -------------
    gemm16_wmma_kernel<128><<<nb_gem, 128, 0, stream>>>(emb, Wc, logits, 64);
    {
        size_t tot = (size_t)N * 64;
        int nb = (int)((tot + 255) / 256);
        bias_act_kernel<false><<<nb, 256, 0, stream>>>(logits, bc, tot, 63);
    }
    softmax_argmax_kernel<<<nb_Nw, 256, 0, stream>>>(logits, soft, hard, N);
}